// NTMHead_65687229825416
// MI455X (gfx1250) — compile-verified
//
#include <hip/hip_runtime.h>
#include <math.h>

typedef __attribute__((ext_vector_type(2))) float v2f;
typedef __attribute__((ext_vector_type(8))) float v8f;

#define W      128
#define NLOC   16384
#define BATCH  64
#define DCTRL  1024
#define TOTAL_OUT 390   // 3*W + 6

// workspace layout (float offsets)
#define WS_KEYN   0                 // 64*128
#define WS_ERASE  (BATCH*W)         // 64*128
#define WS_ADD    (2*BATCH*W)       // 64*128
#define WS_BETA   (3*BATCH*W)       // 64
#define WS_GATE   (WS_BETA+BATCH)   // 64
#define WS_GAMMA  (WS_GATE+BATCH)   // 64
#define WS_SHIFT  (WS_GAMMA+BATCH)  // 64*3
#define WS_MEMN   32768             // 16384*128 normalized memory

__device__ __forceinline__ float softplusf(float x) {
    return (x > 20.f) ? x : log1pf(expf(x));
}
__device__ __forceinline__ float sigmoidf_(float x) {
    return 1.f / (1.f + expf(-x));
}

// ---------------- K1: FC + activations --------------------------------------
__global__ void __launch_bounds__(256) fc_act_kernel(
    const float* __restrict__ cs, const float* __restrict__ fcw,
    const float* __restrict__ fcb, float* __restrict__ ws) {
    __shared__ float s_cs[DCTRL];
    __shared__ float s_raw[TOTAL_OUT];
    __shared__ float s_invn;
    int b = blockIdx.x, t = threadIdx.x;
    for (int j = t; j < DCTRL; j += 256) s_cs[j] = cs[(size_t)b * DCTRL + j];
    __syncthreads();
    for (int o = t; o < TOTAL_OUT; o += 256) {
        const float* wrow = fcw + (size_t)o * DCTRL;
        float acc = fcb[o];
        for (int j = 0; j < DCTRL; ++j) acc = fmaf(s_cs[j], wrow[j], acc);
        s_raw[o] = acc;
    }
    __syncthreads();
    if (t == 0) {
        float ss = 0.f;
        for (int i = 0; i < W; ++i) ss += s_raw[i] * s_raw[i];
        s_invn = 1.f / fmaxf(sqrtf(ss), 1e-12f);
        ws[WS_BETA  + b] = softplusf(s_raw[W]);
        ws[WS_GATE  + b] = sigmoidf_(s_raw[W + 1]);
        ws[WS_GAMMA + b] = softplusf(s_raw[W + 5]) + 1.f;
        float a0 = s_raw[W + 2], a1 = s_raw[W + 3], a2 = s_raw[W + 4];
        float m  = fmaxf(a0, fmaxf(a1, a2));
        float e0 = expf(a0 - m), e1 = expf(a1 - m), e2 = expf(a2 - m);
        float inv = 1.f / (e0 + e1 + e2);
        ws[WS_SHIFT + b * 3 + 0] = e0 * inv;
        ws[WS_SHIFT + b * 3 + 1] = e1 * inv;
        ws[WS_SHIFT + b * 3 + 2] = e2 * inv;
    }
    __syncthreads();
    float invn = s_invn;
    for (int i = t; i < W; i += 256) {
        ws[WS_KEYN  + b * W + i] = s_raw[i] * invn;
        ws[WS_ERASE + b * W + i] = sigmoidf_(s_raw[W + 6 + i]);
        ws[WS_ADD   + b * W + i] = tanhf(s_raw[2 * W + 6 + i]);
    }
}

// ---------------- K2: l2-normalize memory rows (1 wave32 / row) -------------
__global__ void __launch_bounds__(256) memnorm_kernel(
    const float* __restrict__ mem, float* __restrict__ memn) {
    int wave = threadIdx.x >> 5, lane = threadIdx.x & 31;
    int row = blockIdx.x * 8 + wave;
    const float* src = mem + (size_t)row * W;
    float v[4]; float ss = 0.f;
#pragma unroll
    for (int j = 0; j < 4; ++j) { v[j] = src[lane + 32 * j]; ss = fmaf(v[j], v[j], ss); }
#pragma unroll
    for (int off = 16; off >= 1; off >>= 1) ss += __shfl_xor(ss, off, 32);
    float inv = 1.f / fmaxf(sqrtf(ss), 1e-12f);
    float* dst = memn + (size_t)row * W;
#pragma unroll
    for (int j = 0; j < 4; ++j) dst[lane + 32 * j] = v[j] * inv;
}

// ---------------- K3: sim = keyn @ memn^T via f32 WMMA ----------------------
// A: 16x4 f32 per step (2 VGPRs, K split across lane halves), C: v8f per tile.
__global__ void __launch_bounds__(256) sim_wmma_kernel(
    const float* __restrict__ keyn, const float* __restrict__ memn,
    float* __restrict__ sim) {
    __shared__ float s_a[16 * W];   // keyn tile for this m-block (8 KB)
    int t = threadIdx.x;
    int m0 = blockIdx.y * 16;
    for (int i = t; i < 16 * W; i += 256)
        s_a[i] = keyn[(size_t)(m0 + (i >> 7)) * W + (i & (W - 1))];
    __syncthreads();
    int wave = t >> 5, lane = t & 31;
    int n0 = (blockIdx.x * 8 + wave) * 16;
    int mrow = lane & 15, khalf = lane >> 4;
    const float* arow = s_a + mrow * W;
    const float* brow = memn + (size_t)(n0 + mrow) * W;
    v8f c = {};
#pragma unroll
    for (int k0 = 0; k0 < W; k0 += 4) {
        int kk = k0 + 2 * khalf;
        v2f a, bb;
        a.x  = arow[kk]; a.y  = arow[kk + 1];
        bb.x = brow[kk]; bb.y = brow[kk + 1];
        c = __builtin_amdgcn_wmma_f32_16x16x4_f32(false, a, false, bb,
                                                  (short)0, c, false, false);
    }
#pragma unroll
    for (int r = 0; r < 8; ++r) {
        int row = m0 + r + 8 * khalf;          // C: M = r + 8*(lane>=16)
        sim[(size_t)row * NLOC + n0 + mrow] = c[r];
    }
}

// ---------------- block reductions (small LDS only) -------------------------
__device__ __forceinline__ float blockReduceMax(float v, float* red) {
    int lane = threadIdx.x & 31, wid = threadIdx.x >> 5;
#pragma unroll
    for (int o = 16; o >= 1; o >>= 1) v = fmaxf(v, __shfl_xor(v, o, 32));
    if (lane == 0) red[wid] = v;
    __syncthreads();
    float m = (threadIdx.x < 32) ? red[threadIdx.x] : -3.402823e38f;
    if (wid == 0) {
#pragma unroll
        for (int o = 16; o >= 1; o >>= 1) m = fmaxf(m, __shfl_xor(m, o, 32));
        if (lane == 0) red[0] = m;
    }
    __syncthreads();
    float res = red[0];
    __syncthreads();
    return res;
}

__device__ __forceinline__ float blockReduceSum(float v, float* red) {
    int lane = threadIdx.x & 31, wid = threadIdx.x >> 5;
#pragma unroll
    for (int o = 16; o >= 1; o >>= 1) v += __shfl_xor(v, o, 32);
    if (lane == 0) red[wid] = v;
    __syncthreads();
    float s = (threadIdx.x < 32) ? red[threadIdx.x] : 0.f;
    if (wid == 0) {
#pragma unroll
        for (int o = 16; o >= 1; o >>= 1) s += __shfl_xor(s, o, 32);
        if (lane == 0) red[0] = s;
    }
    __syncthreads();
    float res = red[0];
    __syncthreads();
    return res;
}

// ---------------- K4a: softmax(beta*sim) + gate blend, in place -------------
__global__ void __launch_bounds__(1024) softmax_gate_kernel(
    const float* __restrict__ wprev, const float* __restrict__ ws,
    float* __restrict__ io /* logits in, w_g out */) {
    __shared__ float red[32];
    int b = blockIdx.x, t = threadIdx.x;
    float beta = ws[WS_BETA + b], gate = ws[WS_GATE + b];
    float* r = io + (size_t)b * NLOC;
    float lmax = -3.402823e38f;
    for (int i = t; i < NLOC; i += 1024) lmax = fmaxf(lmax, r[i] * beta);
    float gmax = blockReduceMax(lmax, red);
    float lsum = 0.f;
    for (int i = t; i < NLOC; i += 1024) lsum += expf(r[i] * beta - gmax);
    float ginv = 1.f / blockReduceSum(lsum, red);
    const float* wp = wprev + (size_t)b * NLOC;
    float og = 1.f - gate;
    for (int i = t; i < NLOC; i += 1024) {
        float wc = expf(r[i] * beta - gmax) * ginv;
        r[i] = gate * wc + og * wp[i];   // disjoint per-thread indices: safe
    }
}

// ---------------- K4b: circular shift + sharpen + renormalize ---------------
__global__ void __launch_bounds__(1024) sharpen_kernel(
    const float* __restrict__ ws, float* __restrict__ io /* w_g in, w_new out */) {
    __shared__ float red[32];
    int b = blockIdx.x, t = threadIdx.x;
    float s0 = ws[WS_SHIFT + b * 3 + 0];
    float s1 = ws[WS_SHIFT + b * 3 + 1];
    float s2 = ws[WS_SHIFT + b * 3 + 2];
    float gamma = ws[WS_GAMMA + b];
    float* r = io + (size_t)b * NLOC;
    float wpow[NLOC / 1024];
    float psum = 0.f;
    int idx = 0;
    for (int i = t; i < NLOC; i += 1024, ++idx) {
        int im = (i == 0) ? NLOC - 1 : i - 1;
        int ip = (i == NLOC - 1) ? 0 : i + 1;
        float v = s0 * r[im] + s1 * r[i] + s2 * r[ip];
        v = powf(fmaxf(v, 1e-9f), gamma);
        wpow[idx] = v; psum += v;
    }
    // blockReduceSum's barriers separate all reads above from writes below
    float inv = 1.f / (blockReduceSum(psum, red) + 1e-9f);
    idx = 0;
    for (int i = t; i < NLOC; i += 1024, ++idx) r[i] = wpow[idx] * inv;
}

// ---------------- K5: erase/add outer products + memory update (WMMA) -------
__global__ void __launch_bounds__(256) update_wmma_kernel(
    const float* __restrict__ wnew, const float* __restrict__ mem,
    const float* __restrict__ ws, float* __restrict__ newmem) {
    __shared__ float s_w[BATCH * 16];   // w_new[b][n0+j] at [b*16+j] (4 KB)
    int t = threadIdx.x;
    int n0 = blockIdx.x * 16;
    for (int i = t; i < BATCH * 16; i += 256) {
        int b = i >> 4, j = i & 15;
        s_w[i] = wnew[(size_t)b * NLOC + n0 + j];
    }
    __syncthreads();
    int wave = t >> 5, lane = t & 31;
    int w0 = wave * 16;                 // 8 waves cover W=128
    int mrow = lane & 15, khalf = lane >> 4;
    const float* er = ws + WS_ERASE;
    const float* ad = ws + WS_ADD;
    v8f ce = {}, ca = {};
#pragma unroll
    for (int k0 = 0; k0 < BATCH; k0 += 4) {
        int kk = k0 + 2 * khalf;
        v2f a, be, ba;
        a.x  = s_w[kk * 16 + mrow];        a.y  = s_w[(kk + 1) * 16 + mrow];
        be.x = er[kk * W + w0 + mrow];     be.y = er[(kk + 1) * W + w0 + mrow];
        ba.x = ad[kk * W + w0 + mrow];     ba.y = ad[(kk + 1) * W + w0 + mrow];
        ce = __builtin_amdgcn_wmma_f32_16x16x4_f32(false, a, false, be,
                                                   (short)0, ce, false, false);
        ca = __builtin_amdgcn_wmma_f32_16x16x4_f32(false, a, false, ba,
                                                   (short)0, ca, false, false);
    }
    const float invB = 1.f / (float)BATCH;
#pragma unroll
    for (int r = 0; r < 8; ++r) {
        int n = n0 + r + 8 * khalf;
        size_t idx = (size_t)n * W + w0 + mrow;
        float em = ce[r] * invB, am = ca[r] * invB;
        newmem[idx] = mem[idx] * (1.f - em) + am;
    }
}

// ---------------------------------------------------------------------------
extern "C" void kernel_launch(void* const* d_in, const int* in_sizes, int n_in,
                              void* d_out, int out_size, void* d_ws, size_t ws_size,
                              hipStream_t stream) {
    (void)in_sizes; (void)n_in; (void)out_size; (void)ws_size;
    const float* cs    = (const float*)d_in[0];   // (64,1024)
    const float* mem   = (const float*)d_in[1];   // (16384,128)
    const float* wprev = (const float*)d_in[2];   // (64,16384)
    const float* fcw   = (const float*)d_in[3];   // (390,1024)
    const float* fcb   = (const float*)d_in[4];   // (390,)
    float* wsf     = (float*)d_ws;
    float* out_w   = (float*)d_out;                     // w_new: 64*16384
    float* out_mem = out_w + (size_t)BATCH * NLOC;      // new_memory: 16384*128

    fc_act_kernel<<<BATCH, 256, 0, stream>>>(cs, fcw, fcb, wsf);
    memnorm_kernel<<<NLOC / 8, 256, 0, stream>>>(mem, wsf + WS_MEMN);
    dim3 g3(NLOC / (8 * 16), BATCH / 16);               // (128, 4)
    sim_wmma_kernel<<<g3, 256, 0, stream>>>(wsf + WS_KEYN, wsf + WS_MEMN, out_w);
    softmax_gate_kernel<<<BATCH, 1024, 0, stream>>>(wprev, wsf, out_w);
    sharpen_kernel<<<BATCH, 1024, 0, stream>>>(wsf, out_w);
    update_wmma_kernel<<<NLOC / 16, 256, 0, stream>>>(out_w, mem, wsf, out_mem);
}